// TiledCorrelation3D_14705968022014
// MI455X (gfx1250) — compile-verified
//
#include <hip/hip_runtime.h>

typedef float v2f __attribute__((ext_vector_type(2)));
typedef float v8f __attribute__((ext_vector_type(8)));
typedef int   v4i __attribute__((ext_vector_type(4)));

#define TILE    5
#define RADIUS  2
#define DIM     128
#define H       128
#define W       128
#define HW      (H * W)
#define NB      16

#define LDS_STRIDE 132        // 128 + 4 pad -> conflict-free column reads
#define NCOLS      64         // N columns handled per workgroup
#define NTHREADS   128        // 4 waves (wave32)

#if defined(__AMDGCN__) && __has_builtin(__builtin_amdgcn_global_load_async_to_lds_b128)
#define USE_ASYNC_LDS 1
#else
#define USE_ASYNC_LDS 0
#endif

#if USE_ASYNC_LDS
typedef __attribute__((address_space(1))) v4i* g1_v4i;   // global 128-bit ptr
typedef __attribute__((address_space(3))) v4i* l3_v4i;   // LDS 128-bit ptr
#endif

__global__ __launch_bounds__(NTHREADS)
void tiled_corr_wmma_f32(const float* __restrict__ fa,
                         const float* __restrict__ fb,
                         const int*   __restrict__ pts,
                         float*       __restrict__ out)
{
    __shared__ float sA[32 * LDS_STRIDE];      // padded 32 x 128 A tile
    __shared__ float sB[NCOLS * LDS_STRIDE];   // 64 x 128 B chunk

    const int t     = threadIdx.x;
    const int blk   = blockIdx.x;
    const int b     = blk >> 8;                 // 256 chunks per batch
    const int chunk = blk & 255;
    const int n0    = chunk * NCOLS;

    const int py = pts[2 * b + 0] - RADIUS;     // tile top-left (H dim)
    const int px = pts[2 * b + 1] - RADIUS;     // tile top-left (W dim)

    const float* fab = fa + (size_t)b * HW * DIM;
    const float* fbb = fb + (size_t)b * HW * DIM;

#if USE_ASYNC_LDS
    // ---- Stage A tile via async global->LDS (ASYNCcnt path); pad rows with
    //      plain LDS zero stores (DScnt path). Both drained before barrier. ----
    for (int idx = t; idx < 32 * 32; idx += NTHREADS) {
        const int p  = idx >> 5;                // padded row 0..31
        const int d4 = idx & 31;                // 16-byte column
        float* dst = &sA[p * LDS_STRIDE + d4 * 4];
        if (p < TILE * TILE) {
            const int ty = p / TILE, tx = p % TILE;
            const float* src = fab + ((size_t)(py + ty) * W + (px + tx)) * DIM + d4 * 4;
            __builtin_amdgcn_global_load_async_to_lds_b128(
                (g1_v4i)src, (l3_v4i)dst, 0, 0);
        } else {
            dst[0] = 0.f; dst[1] = 0.f; dst[2] = 0.f; dst[3] = 0.f;
        }
    }

    // ---- Stage B chunk via async global->LDS, 16B per lane, coalesced ----
    for (int idx = t; idx < NCOLS * 32; idx += NTHREADS) {
        const int r  = idx >> 5;
        const int d4 = idx & 31;
        const float* src = fbb + (size_t)(n0 + r) * DIM + d4 * 4;
        float*       dst = &sB[r * LDS_STRIDE + d4 * 4];
        __builtin_amdgcn_global_load_async_to_lds_b128(
            (g1_v4i)src, (l3_v4i)dst, 0, 0);
    }

#if __has_builtin(__builtin_amdgcn_s_wait_asynccnt)
    __builtin_amdgcn_s_wait_asynccnt(0);
#else
    asm volatile("s_wait_asynccnt 0x0" ::: "memory");
#endif
#else  // !USE_ASYNC_LDS: global->VGPR->ds_store fallback
    for (int idx = t; idx < 32 * 32; idx += NTHREADS) {
        const int p  = idx >> 5;
        const int d4 = idx & 31;
        float4 val = make_float4(0.f, 0.f, 0.f, 0.f);
        if (p < TILE * TILE) {
            const int ty = p / TILE, tx = p % TILE;
            val = *(const float4*)(fab + ((size_t)(py + ty) * W + (px + tx)) * DIM + d4 * 4);
        }
        float* dst = &sA[p * LDS_STRIDE + d4 * 4];
        dst[0] = val.x; dst[1] = val.y; dst[2] = val.z; dst[3] = val.w;
    }
    for (int idx = t; idx < NCOLS * 32; idx += NTHREADS) {
        const int r  = idx >> 5;
        const int d4 = idx & 31;
        const float4 val = *(const float4*)(fbb + (size_t)(n0 + r) * DIM + d4 * 4);
        float* dst = &sB[r * LDS_STRIDE + d4 * 4];
        dst[0] = val.x; dst[1] = val.y; dst[2] = val.z; dst[3] = val.w;
    }
#endif

    __syncthreads();

    // ---- Per-wave 16x16 output tiles via V_WMMA_F32_16X16X4_F32 ----
    const int wv   = t >> 5;          // wave 0..3 -> 16-column N sub-tile
    const int lane = t & 31;
    const int m    = lane & 15;       // M row (A) / N column (B,C,D)
    const int g2   = (lane >> 4) * 2; // K-pair select: lanes 16..31 hold K=2,3

    const float* pa0 = &sA[(m)      * LDS_STRIDE + g2];
    const float* pa1 = &sA[(m + 16) * LDS_STRIDE + g2];
    const float* pb  = &sB[(wv * 16 + m) * LDS_STRIDE + g2];

    v8f c0 = {0.f, 0.f, 0.f, 0.f, 0.f, 0.f, 0.f, 0.f};   // M rows 0..15
    v8f c1 = {0.f, 0.f, 0.f, 0.f, 0.f, 0.f, 0.f, 0.f};   // M rows 16..31 (24 valid)

#pragma unroll 4
    for (int k = 0; k < DIM; k += 4) {
        const v2f a0 = *(const v2f*)(pa0 + k);
        const v2f a1 = *(const v2f*)(pa1 + k);
        const v2f bb = *(const v2f*)(pb + k);
        c0 = __builtin_amdgcn_wmma_f32_16x16x4_f32(false, a0, false, bb,
                                                   (short)0, c0, false, false);
        c1 = __builtin_amdgcn_wmma_f32_16x16x4_f32(false, a1, false, bb,
                                                   (short)0, c1, false, false);
    }

    // ---- Store: VGPR v holds row M = v + 8*(lane>=16), N = lane&15 ----
    const int  gsel = lane >> 4;
    const int  ncol = n0 + wv * 16 + m;
    float* outb = out + (size_t)b * (TILE * TILE) * HW;

#pragma unroll
    for (int v = 0; v < 8; ++v) {
        const int p0 = v + 8 * gsel;            // 0..15, always valid
        outb[(size_t)p0 * HW + ncol] = c0[v];
    }
    if (gsel == 0) {
        // lanes 0..15: c1 rows 16..23, all valid
#pragma unroll
        for (int v = 0; v < 8; ++v)
            outb[(size_t)(16 + v) * HW + ncol] = c1[v];
    } else {
        // lanes 16..31: only row 24 (v = 0) is inside the 25-row output
        outb[(size_t)24 * HW + ncol] = c1[0];
    }
}

extern "C" void kernel_launch(void* const* d_in, const int* in_sizes, int n_in,
                              void* d_out, int out_size, void* d_ws, size_t ws_size,
                              hipStream_t stream)
{
    (void)in_sizes; (void)n_in; (void)out_size; (void)d_ws; (void)ws_size;
    const float* fa  = (const float*)d_in[0];
    const float* fb  = (const float*)d_in[1];
    const int*   pts = (const int*)d_in[2];
    float*       out = (float*)d_out;

    const int grid = NB * (HW / NCOLS);   // 16 * 256 = 4096 workgroups
    tiled_corr_wmma_f32<<<grid, NTHREADS, 0, stream>>>(fa, fb, pts, out);
}